// PNFCModule_12000138625283
// MI455X (gfx1250) — compile-verified
//
#include <hip/hip_runtime.h>
#include <hip/hip_bf16.h>

typedef float v2f __attribute__((ext_vector_type(2)));
typedef float v8f __attribute__((ext_vector_type(8)));

#define NROWS   200000
#define DTXT    768
#define DIMG    1024
#define DC      256
#define KTOP    10
#define NEG_INF (-3.402823466e+38f)

// ---------------------------------------------------------------------------
// K1: sims[i] = dot(cand_text[i], q) / max(||cand_text[i]||, 1e-12)
// (dropping the positive global 1/||q|| scale does not change top-k order)
// One wave32 per row; float4 coalesced loads; q staged in LDS.
// ---------------------------------------------------------------------------
__global__ void sims_kernel(const float* __restrict__ cand,
                            const float* __restrict__ q,
                            float* __restrict__ sims, int nrows) {
  __shared__ float qs[DTXT];
  for (int i = threadIdx.x; i < DTXT; i += blockDim.x) qs[i] = q[i];
  __syncthreads();
  const int wavesPerBlock = blockDim.x >> 5;
  const int wave = blockIdx.x * wavesPerBlock + (threadIdx.x >> 5);
  const int lane = threadIdx.x & 31;
  if (wave >= nrows) return;
  const float* row = cand + (size_t)wave * DTXT;
  float dot = 0.f, nrm = 0.f;
#pragma unroll
  for (int i = 0; i < 6; ++i) {
    const int base = (lane + 32 * i) * 4;     // 768 = 6 * 32 * 4
    float4 c = *(const float4*)(row + base);
    float4 qq = *(const float4*)(qs + base);
    dot = fmaf(c.x, qq.x, dot); nrm = fmaf(c.x, c.x, nrm);
    dot = fmaf(c.y, qq.y, dot); nrm = fmaf(c.y, c.y, nrm);
    dot = fmaf(c.z, qq.z, dot); nrm = fmaf(c.z, c.z, nrm);
    dot = fmaf(c.w, qq.w, dot); nrm = fmaf(c.w, c.w, nrm);
  }
#pragma unroll
  for (int off = 16; off > 0; off >>= 1) {
    dot += __shfl_xor(dot, off, 32);
    nrm += __shfl_xor(nrm, off, 32);
  }
  if (lane == 0) sims[wave] = dot / fmaxf(sqrtf(nrm), 1e-12f);
}

// ---------------------------------------------------------------------------
// K2: deterministic top-10 (value desc, tie -> lower index) over sims[N].
// ---------------------------------------------------------------------------
#define TKT 512
__global__ void topk_kernel(const float* __restrict__ sims, int n,
                            int* __restrict__ out_idx) {
  __shared__ float svals[TKT * KTOP];
  __shared__ int   sidx [TKT * KTOP];
  __shared__ float rval[TKT];
  __shared__ int   rpos[TKT];
  const int t = threadIdx.x;

  float lv[KTOP]; int li[KTOP];
#pragma unroll
  for (int j = 0; j < KTOP; ++j) { lv[j] = NEG_INF; li[j] = 0x7fffffff; }

  for (int i = t; i < n; i += TKT) {
    float v = sims[i];
    int   ix = i;
#pragma unroll
    for (int j = 0; j < KTOP; ++j) {
      bool better = (v > lv[j]) || (v == lv[j] && ix < li[j]);
      if (better) {
        float tv = lv[j]; int ti = li[j];
        lv[j] = v; li[j] = ix; v = tv; ix = ti;
      }
    }
  }
#pragma unroll
  for (int j = 0; j < KTOP; ++j) { svals[t * KTOP + j] = lv[j]; sidx[t * KTOP + j] = li[j]; }
  __syncthreads();

  const int M = TKT * KTOP;
  for (int round = 0; round < KTOP; ++round) {
    float bv = NEG_INF; int bslot = t; int bidx = 0x7fffffff;
    for (int s = t; s < M; s += TKT) {
      float v = svals[s]; int ix = sidx[s];
      if (v > bv || (v == bv && ix < bidx)) { bv = v; bidx = ix; bslot = s; }
    }
    rval[t] = bv; rpos[t] = bslot;
    __syncthreads();
    for (int stride = TKT / 2; stride > 0; stride >>= 1) {
      if (t < stride) {
        float v1 = rval[t], v2 = rval[t + stride];
        int p1 = rpos[t], p2 = rpos[t + stride];
        int i1 = sidx[p1], i2 = sidx[p2];
        if (v2 > v1 || (v2 == v1 && i2 < i1)) { rval[t] = v2; rpos[t] = p2; }
      }
      __syncthreads();
    }
    if (t == 0) {
      int slot = rpos[0];
      out_idx[round] = sidx[slot];
      svals[slot] = NEG_INF;
    }
    __syncthreads();
  }
}

// ---------------------------------------------------------------------------
// K3: build padded input matrices.
// Xtxt[16,768]: row0=query, rows1..10=cand_text[idx], rest 0
// Ximg[16,1024]: rows1..10=cand_img[idx], rest 0
// ---------------------------------------------------------------------------
__global__ void gather_kernel(const float* __restrict__ query,
                              const float* __restrict__ cand_text,
                              const float* __restrict__ cand_img,
                              const int* __restrict__ idx,
                              float* __restrict__ Xtxt,
                              float* __restrict__ Ximg) {
  const int t = threadIdx.x;
  for (int p = t; p < 16 * DTXT; p += 256) {
    int r = p / DTXT, c = p % DTXT;
    float v = 0.f;
    if (r == 0) v = query[c];
    else if (r <= KTOP) v = cand_text[(size_t)idx[r - 1] * DTXT + c];
    Xtxt[p] = v;
  }
  for (int p = t; p < 16 * DIMG; p += 256) {
    int r = p >> 10, c = p & 1023;
    float v = 0.f;
    if (r >= 1 && r <= KTOP) v = cand_img[(size_t)idx[r - 1] * DIMG + c];
    Ximg[p] = v;
  }
}

// ---------------------------------------------------------------------------
// K4: Y[16,256] = X[16,K] @ W[256,K]^T + bias   via V_WMMA_F32_16X16X4_F32.
// Grid: 16 blocks x 32 threads; one wave owns a 16-wide column tile.
// A layout (ISA 7.12.2, 32-bit 16x4): lanes 0-15 -> K={k,k+1}, 16-31 -> K={k+2,k+3}.
// B mirrored by column; C/D: VGPR r = row (r + 8*half), col = lane&15.
// ---------------------------------------------------------------------------
__global__ void wmma_gemm16(const float* __restrict__ X,
                            const float* __restrict__ W,
                            const float* __restrict__ bias,
                            float* __restrict__ Y, int K) {
  const int tile = blockIdx.x;
  const int lane = threadIdx.x;
  const int half = lane >> 4;
  const int l16  = lane & 15;
  const int col  = tile * 16 + l16;

  v8f c;
  const float bval = bias[col];
#pragma unroll
  for (int r = 0; r < 8; ++r) c[r] = bval;

  const float* xrow = X + (size_t)l16 * K;
  const float* wrow = W + (size_t)col * K;
  for (int k = 0; k < K; k += 4) {
    v2f a, b;
    a.x = xrow[k + 2 * half];
    a.y = xrow[k + 2 * half + 1];
    b.x = wrow[k + 2 * half];
    b.y = wrow[k + 2 * half + 1];
    c = __builtin_amdgcn_wmma_f32_16x16x4_f32(
        /*neg_a=*/false, a, /*neg_b=*/false, b,
        /*c_mod=*/(short)0, c, /*reuse_a=*/false, /*reuse_b=*/false);
  }
#pragma unroll
  for (int r = 0; r < 8; ++r) {
    int row = r + 8 * half;
    Y[row * DC + col] = c[r];
  }
}

// ---------------------------------------------------------------------------
// K5: per-row LayerNorm + ReLU on a [16,256] matrix. 16 waves, wave = row.
// ---------------------------------------------------------------------------
__global__ void lnrelu_kernel(const float* __restrict__ Y,
                              const float* __restrict__ g,
                              const float* __restrict__ be,
                              float* __restrict__ P) {
  const int w = threadIdx.x >> 5, lane = threadIdx.x & 31;
  const float* y = Y + w * DC;
  float vals[8];
  float s = 0.f, s2 = 0.f;
#pragma unroll
  for (int i = 0; i < 8; ++i) {
    float v = y[lane + 32 * i];
    vals[i] = v; s += v; s2 += v * v;
  }
#pragma unroll
  for (int off = 16; off > 0; off >>= 1) {
    s += __shfl_xor(s, off, 32);
    s2 += __shfl_xor(s2, off, 32);
  }
  const float m = s * (1.f / DC);
  const float var = s2 * (1.f / DC) - m * m;
  const float inv = rsqrtf(var + 1e-5f);
#pragma unroll
  for (int i = 0; i < 8; ++i) {
    int cidx = lane + 32 * i;
    float o = (vals[i] - m) * inv * g[cidx] + be[cidx];
    P[w * DC + cidx] = fmaxf(o, 0.f);
  }
}

// ---------------------------------------------------------------------------
// K6: KIn/VIn [16,256]: row0 = proto, rows1..10 = Ptxt/Pimg rows, rest 0.
// ---------------------------------------------------------------------------
__global__ void build_kv_kernel(const float* __restrict__ proto,
                                const float* __restrict__ Ptxt,
                                const float* __restrict__ Pimg,
                                float* __restrict__ KIn,
                                float* __restrict__ VIn) {
  const int t = threadIdx.x;
  for (int p = t; p < 16 * DC; p += 256) {
    int r = p >> 8, cc = p & 255;
    float kv = 0.f, vv = 0.f;
    if (r == 0) { kv = proto[cc]; vv = proto[cc]; }
    else if (r <= KTOP) { kv = Ptxt[p]; vv = Pimg[p]; }
    KIn[p] = kv; VIn[p] = vv;
  }
}

// ---------------------------------------------------------------------------
// K7: attention tail + FFN (single row). 512 threads.
// ---------------------------------------------------------------------------
__global__ void finalize_kernel(const float* __restrict__ Ptxt,
                                const float* __restrict__ kp,
                                const float* __restrict__ vp,
                                const float* __restrict__ Wq, const float* __restrict__ bq,
                                const float* __restrict__ Wo, const float* __restrict__ bo,
                                const float* __restrict__ proto,
                                const float* __restrict__ g1, const float* __restrict__ be1,
                                const float* __restrict__ W1, const float* __restrict__ b1,
                                const float* __restrict__ W2, const float* __restrict__ b2,
                                const float* __restrict__ g2, const float* __restrict__ be2,
                                float* __restrict__ out) {
  __shared__ float af[DC];        // avail_feat = Ptxt row 0
  __shared__ float qp[DC];
  __shared__ float sc[4][16];
  __shared__ float attnw[4][16];
  __shared__ float ctx[DC];
  __shared__ float x1[DC];
  __shared__ float out1[DC];
  __shared__ float h1[4 * DC];
  __shared__ float x2[DC];
  __shared__ float red[512];
  const int t = threadIdx.x;

  if (t < DC) af[t] = Ptxt[t];
  __syncthreads();

  // qp = af @ Wq^T + bq
  if (t < DC) {
    float acc = bq[t];
    const float* wr = Wq + t * DC;
    for (int j = 0; j < DC; ++j) acc = fmaf(af[j], wr[j], acc);
    qp[t] = acc;
  }
  __syncthreads();

  // scores[h][n] = dot64(qp_h, kp[n]_h) / 8
  if (t < 44) {
    int h = t / 11, n = t % 11;
    float acc = 0.f;
    const float* kr = kp + n * DC + h * 64;
    const float* qr = qp + h * 64;
    for (int d = 0; d < 64; ++d) acc = fmaf(qr[d], kr[d], acc);
    sc[h][n] = acc * 0.125f;
  }
  __syncthreads();

  // softmax per head over 11 entries
  if (t < 4) {
    float mx = NEG_INF;
    for (int n = 0; n < 11; ++n) mx = fmaxf(mx, sc[t][n]);
    float sm = 0.f;
    for (int n = 0; n < 11; ++n) { float e = __expf(sc[t][n] - mx); attnw[t][n] = e; sm += e; }
    float inv = 1.f / sm;
    for (int n = 0; n < 11; ++n) attnw[t][n] *= inv;
  }
  __syncthreads();

  // ctx = attn @ vp
  if (t < DC) {
    int h = t >> 6;
    float acc = 0.f;
    for (int n = 0; n < 11; ++n) acc = fmaf(attnw[h][n], vp[n * DC + t], acc);
    ctx[t] = acc;
  }
  __syncthreads();

  // x1 = ctx @ Wo^T + bo + proto
  if (t < DC) {
    float acc = bo[t];
    const float* wr = Wo + t * DC;
    for (int j = 0; j < DC; ++j) acc = fmaf(ctx[j], wr[j], acc);
    x1[t] = acc + proto[t];
  }
  __syncthreads();

  // LayerNorm 1
  red[t] = (t < DC) ? x1[t] : 0.f;
  __syncthreads();
  for (int s = 256; s > 0; s >>= 1) { if (t < s) red[t] += red[t + s]; __syncthreads(); }
  const float m1 = red[0] * (1.f / DC);
  __syncthreads();
  { float d = (t < DC) ? (x1[t] - m1) : 0.f; red[t] = d * d; }
  __syncthreads();
  for (int s = 256; s > 0; s >>= 1) { if (t < s) red[t] += red[t + s]; __syncthreads(); }
  const float inv1 = rsqrtf(red[0] * (1.f / DC) + 1e-5f);
  __syncthreads();
  if (t < DC) out1[t] = (x1[t] - m1) * inv1 * g1[t] + be1[t];
  __syncthreads();

  // FFN: h1 = relu(out1 @ W1^T + b1)
  for (int i = t; i < 4 * DC; i += 512) {
    float acc = b1[i];
    const float* wr = W1 + (size_t)i * DC;
    for (int j = 0; j < DC; ++j) acc = fmaf(out1[j], wr[j], acc);
    h1[i] = fmaxf(acc, 0.f);
  }
  __syncthreads();

  // x2 = h1 @ W2^T + b2 + out1
  if (t < DC) {
    float acc = b2[t];
    const float* wr = W2 + (size_t)t * (4 * DC);
    for (int j = 0; j < 4 * DC; ++j) acc = fmaf(h1[j], wr[j], acc);
    x2[t] = acc + out1[t];
  }
  __syncthreads();

  // LayerNorm 2 + proto
  red[t] = (t < DC) ? x2[t] : 0.f;
  __syncthreads();
  for (int s = 256; s > 0; s >>= 1) { if (t < s) red[t] += red[t + s]; __syncthreads(); }
  const float m2 = red[0] * (1.f / DC);
  __syncthreads();
  { float d = (t < DC) ? (x2[t] - m2) : 0.f; red[t] = d * d; }
  __syncthreads();
  for (int s = 256; s > 0; s >>= 1) { if (t < s) red[t] += red[t + s]; __syncthreads(); }
  const float inv2 = rsqrtf(red[0] * (1.f / DC) + 1e-5f);
  __syncthreads();
  if (t < DC) out[t] = (x2[t] - m2) * inv2 * g2[t] + be2[t] + proto[t];
}

// ---------------------------------------------------------------------------
extern "C" void kernel_launch(void* const* d_in, const int* in_sizes, int n_in,
                              void* d_out, int out_size, void* d_ws, size_t ws_size,
                              hipStream_t stream) {
  const float* query     = (const float*)d_in[0];
  const float* cand_text = (const float*)d_in[1];
  const float* cand_img  = (const float*)d_in[2];
  const float* proto     = (const float*)d_in[3];
  const float* W_txt = (const float*)d_in[4];
  const float* b_txt = (const float*)d_in[5];
  const float* g_txt = (const float*)d_in[6];
  const float* be_txt = (const float*)d_in[7];
  const float* W_img = (const float*)d_in[8];
  const float* b_img = (const float*)d_in[9];
  const float* g_img = (const float*)d_in[10];
  const float* be_img = (const float*)d_in[11];
  const float* Wq = (const float*)d_in[12]; const float* bq = (const float*)d_in[13];
  const float* Wk = (const float*)d_in[14]; const float* bk = (const float*)d_in[15];
  const float* Wv = (const float*)d_in[16]; const float* bv = (const float*)d_in[17];
  const float* Wo = (const float*)d_in[18]; const float* bo = (const float*)d_in[19];
  const float* g1 = (const float*)d_in[20]; const float* be1 = (const float*)d_in[21];
  const float* W1 = (const float*)d_in[22]; const float* b1 = (const float*)d_in[23];
  const float* W2 = (const float*)d_in[24]; const float* b2 = (const float*)d_in[25];
  const float* g2 = (const float*)d_in[26]; const float* be2 = (const float*)d_in[27];
  const int nrows = in_sizes[1] / DTXT;   // 200000

  // workspace layout (floats)
  float* W  = (float*)d_ws;
  float* sims = W;                          // [200000]
  int*   idx  = (int*)(W + NROWS);          // [16]
  float* F    = W + NROWS + 16;
  float* Xtxt = F;                          // [16*768]
  float* Ximg = Xtxt + 16 * DTXT;           // [16*1024]
  float* Ytxt = Ximg + 16 * DIMG;           // [16*256]
  float* Yimg = Ytxt + 16 * DC;
  float* Ptxt = Yimg + 16 * DC;
  float* Pimg = Ptxt + 16 * DC;
  float* KIn  = Pimg + 16 * DC;
  float* VIn  = KIn + 16 * DC;
  float* kp   = VIn + 16 * DC;
  float* vp   = kp + 16 * DC;

  const int wavesPerBlock = 8;              // 256 threads
  const int nblocks = (nrows + wavesPerBlock - 1) / wavesPerBlock;

  sims_kernel<<<nblocks, 256, 0, stream>>>(cand_text, query, sims, nrows);
  topk_kernel<<<1, TKT, 0, stream>>>(sims, nrows, idx);
  gather_kernel<<<1, 256, 0, stream>>>(query, cand_text, cand_img, idx, Xtxt, Ximg);
  wmma_gemm16<<<16, 32, 0, stream>>>(Xtxt, W_txt, b_txt, Ytxt, DTXT);
  wmma_gemm16<<<16, 32, 0, stream>>>(Ximg, W_img, b_img, Yimg, DIMG);
  lnrelu_kernel<<<1, 512, 0, stream>>>(Ytxt, g_txt, be_txt, Ptxt);
  lnrelu_kernel<<<1, 512, 0, stream>>>(Yimg, g_img, be_img, Pimg);
  build_kv_kernel<<<1, 256, 0, stream>>>(proto, Ptxt, Pimg, KIn, VIn);
  wmma_gemm16<<<16, 32, 0, stream>>>(KIn, Wk, bk, kp, DC);
  wmma_gemm16<<<16, 32, 0, stream>>>(VIn, Wv, bv, vp, DC);
  finalize_kernel<<<1, 512, 0, stream>>>(Ptxt, kp, vp, Wq, bq, Wo, bo, proto,
                                         g1, be1, W1, b1, W2, b2, g2, be2,
                                         (float*)d_out);
}